// NonLocalMeans_2156073583166
// MI455X (gfx1250) — compile-verified
//
#include <hip/hip_runtime.h>
#include <math.h>

// ---------------------------------------------------------------------------
// Non-Local Means (per-channel) on gfx1250.
//   - 58x58 halo tile of one (batch,channel) plane staged in LDS per block
//   - per wave: 16x16 output tile
//   - 7x7 circular box-sum of squared diffs done as two banded matmuls using
//     V_WMMA_F32_16X16X4_F32 (K padded 22->24, 6 chained K=4 steps per pass)
//   - raw v_sqrt_f32 / v_exp_f32 / v_rcp_f32 for the weight math
// ---------------------------------------------------------------------------

typedef float v2f __attribute__((ext_vector_type(2)));
typedef float v8f __attribute__((ext_vector_type(8)));

#define H_IMG    512
#define W_IMG    512
#define SEARCH_R 10
#define PATCH_R  3
#define TILE     32
#define HALO     (SEARCH_R + PATCH_R)   /* 13 */
#define TDIM     (TILE + 2 * HALO)      /* 58 */
#define TSTRIDE  88                     /* row-pair (delta 2) -> 2*88%64=48: bank-disjoint */
#define TS_STRIDE 35                    /* 35^-1 mod 64 = 11 -> conflict-free A-column walk */

#define LOG2E 1.44269504088896340736f

__global__ __launch_bounds__(128)
void nlm_wmma_kernel(const float* __restrict__ img,
                     const float* __restrict__ hptr,
                     float* __restrict__ out)
{
    __shared__ float tile[TDIM * TSTRIDE];          // 20.4 KB
    __shared__ float tsc[4][16 * TS_STRIDE];        // per-wave 16x35 scratch (9 KB)

    const int tid   = threadIdx.x;
    const int lane  = tid & 31;
    const int wv    = tid >> 5;          // wave id 0..3
    const int bx    = blockIdx.x;
    const int by    = blockIdx.y;
    const int plane = blockIdx.z;        // b*3 + c, 12 planes

    const float* __restrict__ pimg = img + (size_t)plane * H_IMG * W_IMG;

    // ---- cooperative halo tile load with circular wrap (512 is pow2) ----
    const int y0 = by * TILE - HALO;
    const int x0 = bx * TILE - HALO;
    for (int i = tid; i < TDIM * TDIM; i += 128) {
        int r  = i / TDIM;
        int c  = i - r * TDIM;
        int gy = (y0 + r) & (H_IMG - 1);
        int gx = (x0 + c) & (W_IMG - 1);
        tile[r * TSTRIDE + c] = pimg[gy * W_IMG + gx];
    }
    __syncthreads();

    const float hv   = hptr[0];
    const float hp   = fmaxf(hv, 0.0f) + 1e-8f;
    const float kexp = LOG2E / hp;       // exp(-d/h) == exp2(-d * log2e/h)

    const int subY = (wv >> 1) * 16;     // wave sub-tile origin in block
    const int subX = (wv & 1) * 16;
    const int hi   = lane >> 4;          // lane-half (selects K offset 0 / 2)
    const int ln   = lane & 15;

    // Band constants (0/1, width-7 band, K in [0,22)).
    //  Vertical pass uses them as A:  A[m][k] = (m <= k <= m+6)
    //  Horizontal pass uses them as B: B[k][n] = (n <= k <= n+6)
    //  Identical per-lane values: lane%16 = m (resp. n), VGPR j holds K=4c+2*hi+j.
    v2f band[6];
#pragma unroll
    for (int c = 0; c < 6; ++c) {
        int k0 = 4 * c + 2 * hi;
#pragma unroll
        for (int j = 0; j < 2; ++j) {
            int k = k0 + j;
            band[c][j] = (k >= ln && k <= ln + 6 && k < 22) ? 1.0f : 0.0f;
        }
    }

    // lane masks replacing per-element branches (0/1 floats)
    const float qm1 = (ln < 6) ? 1.0f : 0.0f;   // col-group 1: D col = 16+ln < 22
    const float rm5 = (hi == 0) ? 1.0f : 0.0f;  // K-chunk 5: rows 20,21 ok / 22,23 pad

    float den[8], ws[8];
#pragma unroll
    for (int v = 0; v < 8; ++v) { den[v] = 0.0f; ws[v] = 0.0f; }

    float* __restrict__ tw = &tsc[wv][0];

    // D-region: rows r=0..21 <-> LDS row subY+10+r, cols q=0..21 <-> subX+10+q.
    // Per-lane, per-colgroup base pointers; element (c,j) lives at +(4c+j)*TSTRIDE.
    const int cb0 = (subY + 10 + 2 * hi) * TSTRIDE + (subX + 10) + ln;
    const float* __restrict__ pg[2] = { &tile[cb0], &tile[cb0 + 16] };

#pragma unroll 1
    for (int sy = -SEARCH_R; sy <= SEARCH_R; ++sy) {
#pragma unroll 1
        for (int sx = -SEARCH_R; sx <= SEARCH_R; ++sx) {
            const int soff = -sy * TSTRIDE - sx;   // offset to shifted sample

            // ---- pass 1: T(16x22) = band(16x24) x D(24x22), two col-groups ----
            v8f tacc[2];
#pragma unroll
            for (int g = 0; g < 2; ++g) {
                const float* __restrict__ p  = pg[g];
                const float* __restrict__ ps = pg[g] + soff;
                v8f acc = {0.f, 0.f, 0.f, 0.f, 0.f, 0.f, 0.f, 0.f};
#pragma unroll
                for (int c = 0; c < 6; ++c) {
                    v2f bD;
#pragma unroll
                    for (int j = 0; j < 2; ++j) {
                        const int off = (4 * c + j) * TSTRIDE;  // immediate ds offset
                        float d = p[off] - ps[off];
                        if (c == 5) d *= rm5;                   // pad rows 22,23 -> 0
                        if (g == 1) d *= qm1;                   // pad cols 22..31 -> 0
                        bD[j] = d * d;
                    }
                    acc = __builtin_amdgcn_wmma_f32_16x16x4_f32(
                        false, band[c], false, bD, (short)0, acc, false, false);
                }
                tacc[g] = acc;
            }

            // C layout -> LDS scratch (row m = v + 8*hi, col = lane%16 [+16]).
            // tacc[1] is exactly 0 for ln>=6, so cols 22..31 (incl. K-pad 22,23)
            // are auto-zeroed: pass-2 A loads need no bounds check.
#pragma unroll
            for (int v = 0; v < 8; ++v) {
                const int m = v + 8 * hi;
                tw[m * TS_STRIDE + ln]      = tacc[0][v];
                tw[m * TS_STRIDE + 16 + ln] = tacc[1][v];
            }

            // ---- pass 2: Out(16x16) = T(16x24) x band(24x16) ----
            v8f oacc = {0.f, 0.f, 0.f, 0.f, 0.f, 0.f, 0.f, 0.f};
            const int baseA = ln * TS_STRIDE + 2 * hi;   // A row M = lane%16
#pragma unroll
            for (int c = 0; c < 6; ++c) {
                v2f aT;
                aT[0] = tw[baseA + 4 * c];
                aT[1] = tw[baseA + 4 * c + 1];
                oacc = __builtin_amdgcn_wmma_f32_16x16x4_f32(
                    false, aT, false, band[c], (short)0, oacc, false, false);
            }

            // ---- epilogue: weight + accumulate (raw TRANS ops, overlap WMMA) ----
            const int sbase = (HALO + subY - sy) * TSTRIDE + (HALO + subX - sx) + ln;
#pragma unroll
            for (int v = 0; v < 8; ++v) {
                const int m   = v + 8 * hi;
                const float u = oacc[v];                    // box-summed sq-diff >= 0
                const float dist = __builtin_amdgcn_sqrtf(u);
                const float w    = __builtin_amdgcn_exp2f(-dist * kexp);
                const float sval = tile[sbase + m * TSTRIDE];
                den[v] += sval * w;
                ws[v]  += w;
            }
        }
    }

    // ---- write back: clip(den/wsum, 0, 1) ----
    float* __restrict__ pout = out + (size_t)plane * H_IMG * W_IMG;
    const int gy0 = by * TILE + subY;
    const int gx  = bx * TILE + subX + ln;
#pragma unroll
    for (int v = 0; v < 8; ++v) {
        const int m = v + 8 * hi;
        float r = den[v] * __builtin_amdgcn_rcpf(ws[v]);
        r = fminf(fmaxf(r, 0.0f), 1.0f);
        pout[(size_t)(gy0 + m) * W_IMG + gx] = r;
    }
}

extern "C" void kernel_launch(void* const* d_in, const int* in_sizes, int n_in,
                              void* d_out, int out_size, void* d_ws, size_t ws_size,
                              hipStream_t stream)
{
    (void)in_sizes; (void)n_in; (void)d_ws; (void)ws_size; (void)out_size;
    const float* rgb = (const float*)d_in[0];   // [4,3,512,512] fp32
    const float* h   = (const float*)d_in[1];   // [1] fp32
    float* out       = (float*)d_out;           // [4,3,512,512] fp32

    dim3 grid(W_IMG / TILE, H_IMG / TILE, 4 * 3);  // 16 x 16 x 12
    dim3 block(128, 1, 1);                         // 4 waves of 32
    nlm_wmma_kernel<<<grid, block, 0, stream>>>(rgb, h, out);
}